// TjlNet_53334903882348
// MI455X (gfx1250) — compile-verified
//
#include <hip/hip_runtime.h>
#include <hip/hip_bf16.h>

// ---------------------------------------------------------------------------
// 3-layer GIN on MI455X (gfx1250): scatter-add aggregation + fused 2xGEMM MLP
// using V_WMMA_F32_16X16X4_F32 (full FP32 precision), batchnorm via global
// column reductions. Workload is HBM/atomic bound (~3 GB traffic, ~39 GFLOP),
// so FP32 WMMA is the right precision: matrix pipe utilization without any
// accuracy loss.
// ---------------------------------------------------------------------------

typedef __attribute__((ext_vector_type(2))) float v2f;
typedef __attribute__((ext_vector_type(8))) float v8f;

constexpr int DIMC = 128;   // feature dim
constexpr int LDA  = 132;   // LDS row stride (floats): 4-dword skew -> no bank conflicts
constexpr int BM   = 64;    // rows per block in the MLP kernel

__device__ __forceinline__ void atomAddF(float* p, float v) {
  __hip_atomic_fetch_add(p, v, __ATOMIC_RELAXED, __HIP_MEMORY_SCOPE_AGENT);
}
__device__ __forceinline__ void atomAddLds(float* p, float v) {
  __hip_atomic_fetch_add(p, v, __ATOMIC_RELAXED, __HIP_MEMORY_SCOPE_WORKGROUP);
}

// ------------------------------- utility kernels ---------------------------

__global__ __launch_bounds__(256) void copy4_kernel(const float* __restrict__ in,
                                                    float* __restrict__ out, int n4) {
  int i = blockIdx.x * blockDim.x + threadIdx.x;
  if (i < n4) ((float4*)out)[i] = ((const float4*)in)[i];
}

__global__ void zero_sums_kernel(float* __restrict__ cs, float* __restrict__ css) {
  int t = threadIdx.x;
  if (t < DIMC) cs[t] = 0.f;
  else if (t < 2 * DIMC) css[t - DIMC] = 0.f;
}

// --------------------------- edge scatter-add ------------------------------
// One wave per edge: 32 lanes x float4 = 128 features. agg must already hold h
// (so the result is segment_sum + h, i.e. (1+eps)*h with eps=0).

__global__ __launch_bounds__(256) void scatter_kernel(const float* __restrict__ h,
                                                      const int* __restrict__ src,
                                                      const int* __restrict__ dst,
                                                      float* __restrict__ agg, int E) {
  int lane = threadIdx.x & 31;
  int wid  = (blockIdx.x * blockDim.x + threadIdx.x) >> 5;
  int nw   = (gridDim.x * blockDim.x) >> 5;
  for (int e = wid; e < E; e += nw) {
    int s = src[e];
    int d = dst[e];
    float4 v = *(const float4*)(h + (size_t)s * DIMC + lane * 4);
    float* p = agg + (size_t)d * DIMC + lane * 4;
    atomAddF(p + 0, v.x);
    atomAddF(p + 1, v.y);
    atomAddF(p + 2, v.z);
    atomAddF(p + 3, v.w);
  }
}

// ------------------------ fused MLP (GEMM1+ReLU+GEMM2) ---------------------
// Block: 256 threads = 8 waves, tile = 64 rows x 128 cols.
// Wave w: M-tile mt = w>>1 (16 rows), N-half nh = w&1 (64 cols = 4 WMMA tiles).
// FP32 WMMA 16x16x4: A(16x4) lane = m + 16*khalf, vgpr j = K parity;
// B(4x16) lane = n + 16*khalf, vgpr j = K parity; C/D row = v + 8*khalf.

__global__ __launch_bounds__(256) void mlp_kernel(const float* __restrict__ agg,
                                                  const float* __restrict__ W1,
                                                  const float* __restrict__ b1,
                                                  const float* __restrict__ W2,
                                                  const float* __restrict__ b2,
                                                  float* __restrict__ y2,
                                                  float* __restrict__ g_cs,
                                                  float* __restrict__ g_css, int N) {
  __shared__ float lds_x[BM * LDA];   // X tile, then Y1 tile (reused)
  __shared__ float lds_cs[DIMC];
  __shared__ float lds_css[DIMC];

  const int t    = threadIdx.x;
  const int lane = t & 31;
  const int wv   = t >> 5;
  const int mt   = wv >> 1;        // 0..3
  const int nh   = wv & 1;         // 0..1
  const int m    = lane & 15;
  const int kh   = lane >> 4;      // 0 or 1
  const int n0   = nh * 64;
  const int base = blockIdx.x * BM;

  if (t < DIMC) { lds_cs[t] = 0.f; lds_css[t] = 0.f; }

  // --- stage X tile (64 x 128) into LDS, zero-fill past N ---
  {
    const int fc = (t & 31) * 4;
    const int r0 = t >> 5;
#pragma unroll
    for (int i = 0; i < 8; ++i) {
      int r  = r0 + i * 8;
      int gr = base + r;
      float4 v = make_float4(0.f, 0.f, 0.f, 0.f);
      if (gr < N) v = *(const float4*)(agg + (size_t)gr * DIMC + fc);
      *(float4*)(lds_x + r * LDA + fc) = v;
    }
  }
  __syncthreads();

  v8f acc[4];
  const v8f vzero = {};
#pragma unroll
  for (int nt = 0; nt < 4; ++nt) acc[nt] = vzero;

  // --- GEMM1: Y1 = X @ W1 ---
  for (int kk = 0; kk < 32; ++kk) {
    const int k = 4 * kk + 2 * kh;
    v2f a = *(const v2f*)(lds_x + (mt * 16 + m) * LDA + k);
#pragma unroll
    for (int nt = 0; nt < 4; ++nt) {
      const int c = n0 + nt * 16 + m;
      v2f b;
      b.x = W1[k * DIMC + c];
      b.y = W1[(k + 1) * DIMC + c];
      acc[nt] = __builtin_amdgcn_wmma_f32_16x16x4_f32(false, a, false, b,
                                                      (short)0, acc[nt], false, false);
    }
  }
  __syncthreads();   // everyone done reading X

  // --- bias + ReLU, restage Y1 into the same LDS ---
#pragma unroll
  for (int nt = 0; nt < 4; ++nt) {
    const int c  = n0 + nt * 16 + m;
    const float bb = b1[c];
#pragma unroll
    for (int v = 0; v < 8; ++v) {
      int r = mt * 16 + v + 8 * kh;
      float val = fmaxf(acc[nt][v] + bb, 0.f);
      lds_x[r * LDA + c] = val;
    }
  }
  __syncthreads();

#pragma unroll
  for (int nt = 0; nt < 4; ++nt) acc[nt] = vzero;

  // --- GEMM2: Y2 = Y1 @ W2 ---
  for (int kk = 0; kk < 32; ++kk) {
    const int k = 4 * kk + 2 * kh;
    v2f a = *(const v2f*)(lds_x + (mt * 16 + m) * LDA + k);
#pragma unroll
    for (int nt = 0; nt < 4; ++nt) {
      const int c = n0 + nt * 16 + m;
      v2f b;
      b.x = W2[k * DIMC + c];
      b.y = W2[(k + 1) * DIMC + c];
      acc[nt] = __builtin_amdgcn_wmma_f32_16x16x4_f32(false, a, false, b,
                                                      (short)0, acc[nt], false, false);
    }
  }

  // --- bias + ReLU, store y2, accumulate column sum / sumsq ---
#pragma unroll
  for (int nt = 0; nt < 4; ++nt) {
    const int c  = n0 + nt * 16 + m;
    const float bb = b2[c];
    float s = 0.f, ss = 0.f;
#pragma unroll
    for (int v = 0; v < 8; ++v) {
      int gr = base + mt * 16 + v + 8 * kh;
      float val = fmaxf(acc[nt][v] + bb, 0.f);
      if (gr < N) {
        y2[(size_t)gr * DIMC + c] = val;
        s  += val;
        ss += val * val;
      }
    }
    atomAddLds(&lds_cs[c], s);
    atomAddLds(&lds_css[c], ss);
  }
  __syncthreads();

  if (t < DIMC) {
    atomAddF(g_cs  + t, lds_cs[t]);
    atomAddF(g_css + t, lds_css[t]);
  }
}

// --------------------------- batchnorm finalize ----------------------------

__global__ void bn_finalize_kernel(const float* __restrict__ cs, const float* __restrict__ css,
                                   const float* __restrict__ gamma, const float* __restrict__ beta,
                                   float* __restrict__ scsh, float invN) {
  int c = threadIdx.x;          // 128 threads
  float mu  = cs[c] * invN;
  float var = fmaxf(css[c] * invN - mu * mu, 0.f);
  float rs  = rsqrtf(var + 1e-5f);
  float sc  = gamma[c] * rs;
  scsh[c]        = sc;
  scsh[DIMC + c] = beta[c] - mu * sc;
}

// ------------------- batchnorm apply + prep next layer ---------------------
// Reads y2 (in h), writes: out slice, and (if not last layer) h and agg.
// h may alias y2 (element-wise in place).

__global__ __launch_bounds__(256) void bn_apply_kernel(const float* __restrict__ y2,
                                                       const float* __restrict__ scsh,
                                                       float* __restrict__ out,
                                                       float* __restrict__ h,
                                                       float* __restrict__ agg,
                                                       int N, int need_next) {
  int idx = blockIdx.x * blockDim.x + threadIdx.x;   // one float4 each
  if (idx >= N * (DIMC / 4)) return;
  int row = idx >> 5;
  int c4  = (idx & 31) * 4;
  float4 y  = *(const float4*)(y2 + (size_t)row * DIMC + c4);
  float4 sc = *(const float4*)(scsh + c4);
  float4 sh = *(const float4*)(scsh + DIMC + c4);
  float4 z;
  z.x = y.x * sc.x + sh.x;
  z.y = y.y * sc.y + sh.y;
  z.z = y.z * sc.z + sh.z;
  z.w = y.w * sc.w + sh.w;
  *(float4*)(out + (size_t)row * (3 * DIMC) + c4) = z;
  if (need_next) {
    *(float4*)(h   + (size_t)row * DIMC + c4) = z;
    *(float4*)(agg + (size_t)row * DIMC + c4) = z;
  }
}

// ------------------------------- launcher ----------------------------------

extern "C" void kernel_launch(void* const* d_in, const int* in_sizes, int n_in,
                              void* d_out, int out_size, void* d_ws, size_t ws_size,
                              hipStream_t stream) {
  const float* x     = (const float*)d_in[0];
  const int*   ei    = (const int*)d_in[1];
  const float* W1    = (const float*)d_in[2];
  const float* b1    = (const float*)d_in[3];
  const float* W2    = (const float*)d_in[4];
  const float* b2    = (const float*)d_in[5];
  const float* gamma = (const float*)d_in[6];
  const float* beta  = (const float*)d_in[7];
  float* out = (float*)d_out;

  const int N = in_sizes[0] / DIMC;
  const int E = in_sizes[1] / 2;
  const int* src = ei;
  const int* dst = ei + E;

  float* agg  = (float*)d_ws;
  float* h    = agg + (size_t)N * DIMC;
  float* cs   = h + (size_t)N * DIMC;
  float* css  = cs + DIMC;
  float* scsh = css + DIMC;

  const int n4         = N * (DIMC / 4);
  const int copyBlocks = (n4 + 255) / 256;
  const int scatBlocks = (E + 7) / 8;          // 8 waves per block, 1 edge per wave
  const int mlpBlocks  = (N + BM - 1) / BM;
  const int bnBlocks   = (N * (DIMC / 4) + 255) / 256;
  const float invN     = 1.0f / (float)N;

  for (int l = 0; l < 3; ++l) {
    const float* hin = (l == 0) ? x : h;
    if (l == 0)
      copy4_kernel<<<copyBlocks, 256, 0, stream>>>(x, agg, n4);
    zero_sums_kernel<<<1, 256, 0, stream>>>(cs, css);
    scatter_kernel<<<scatBlocks, 256, 0, stream>>>(hin, src, dst, agg, E);
    mlp_kernel<<<mlpBlocks, 256, 0, stream>>>(agg,
                                              W1 + (size_t)l * DIMC * DIMC, b1 + l * DIMC,
                                              W2 + (size_t)l * DIMC * DIMC, b2 + l * DIMC,
                                              h, cs, css, N);
    bn_finalize_kernel<<<1, DIMC, 0, stream>>>(cs, css, gamma + l * DIMC, beta + l * DIMC,
                                               scsh, invN);
    bn_apply_kernel<<<bnBlocks, 256, 0, stream>>>(h, scsh, out + l * DIMC, h, agg,
                                                  N, (l < 2) ? 1 : 0);
  }
}